// Subtractor8Bit_23407571764121
// MI455X (gfx1250) — compile-verified
//
#include <hip/hip_runtime.h>
#include <cstdint>
#include <cstddef>

typedef float v4f __attribute__((ext_vector_type(4)));

#define TPB 256
#define TILES_PER_BLOCK 8

// Async global->LDS 16-byte copy (CDNA5 path, tracked by ASYNCcnt).
// LDS destination address comes from lds_addr, global source address from gaddr.
__device__ __forceinline__ void async_load16(uint32_t lds_addr, const void* gaddr) {
  asm volatile("global_load_async_to_lds_b128 %0, %1, off"
               :: "v"(lds_addr), "v"(gaddr)
               : "memory");
}

__device__ __forceinline__ void wait_async_le4() {
  asm volatile("s_wait_asynccnt 4" ::: "memory");
}
__device__ __forceinline__ void wait_async_0() {
  asm volatile("s_wait_asynccnt 0" ::: "memory");
}

// Pack 8 exact-0/1 floats (index 0 = MSB ... index 7 = LSB) into an int.
__device__ __forceinline__ int pack8(v4f lo, v4f hi) {
  float f = fmaf(lo.x, 128.f,
            fmaf(lo.y,  64.f,
            fmaf(lo.z,  32.f,
            fmaf(lo.w,  16.f,
            fmaf(hi.x,   8.f,
            fmaf(hi.y,   4.f,
            fmaf(hi.z,   2.f, hi.w)))))));
  return (int)f;
}

__device__ __forceinline__ void compute_store(int ua, int ub, int ubin,
                                              v4f* __restrict__ outD,
                                              float* __restrict__ outB,
                                              size_t row) {
  int d = ua - ub - ubin;           // 8-bit subtract with borrow-in
  float borrow = (d < 0) ? 1.0f : 0.0f;
  int m = d & 255;                  // two's-complement low 8 bits == ripple result
  v4f d0, d1;
  d0.x = (float)((m >> 7) & 1);     // column 0 = bit 7 (MSB)
  d0.y = (float)((m >> 6) & 1);
  d0.z = (float)((m >> 5) & 1);
  d0.w = (float)((m >> 4) & 1);
  d1.x = (float)((m >> 3) & 1);
  d1.y = (float)((m >> 2) & 1);
  d1.z = (float)((m >> 1) & 1);
  d1.w = (float)( m       & 1);     // column 7 = bit 0 (LSB)
  __builtin_nontemporal_store(d0, &outD[2 * row]);
  __builtin_nontemporal_store(d1, &outD[2 * row + 1]);
  __builtin_nontemporal_store(borrow, &outB[row]);
}

__global__ void __launch_bounds__(TPB)
sub8_kernel(const float* __restrict__ A, const float* __restrict__ B,
            const float* __restrict__ Bin, float* __restrict__ out,
            int nRows, int nTiles) {
  const int t = threadIdx.x;
  // [buffer][chunk: A-lo, A-hi, B-lo, B-hi][thread] ; 32 KB total
  __shared__ v4f smem[2][4][TPB];
  const uint32_t ldsBase = (uint32_t)(uintptr_t)&smem[0][0][0];

  const v4f* Av = (const v4f*)A;
  const v4f* Bv = (const v4f*)B;
  v4f* outD = (v4f*)out;
  float* outB = out + (size_t)nRows * 8;

  const int tile0 = (int)blockIdx.x * TILES_PER_BLOCK;
  const int tEnd  = min(tile0 + TILES_PER_BLOCK, nTiles);
  const int nT    = tEnd - tile0;

  if (nT > 0) {
    // Prologue: stage tile0 into buffer 0.
    {
      size_t row = (size_t)tile0 * TPB + t;
      uint32_t lb = ldsBase + (uint32_t)t * 16u;
      async_load16(lb +     0u, &Av[2 * row]);
      async_load16(lb +  4096u, &Av[2 * row + 1]);
      async_load16(lb +  8192u, &Bv[2 * row]);
      async_load16(lb + 12288u, &Bv[2 * row + 1]);
    }
    for (int i = 0; i < nT; ++i) {
      const int buf = i & 1;
      if (i + 1 < nT) {
        // Stage next tile into the other buffer, then wait for current tile
        // (4 outstanding ops allowed = the ones just issued).
        size_t row = (size_t)(tile0 + i + 1) * TPB + t;
        uint32_t lb = ldsBase + (uint32_t)((i + 1) & 1) * 16384u
                              + (uint32_t)t * 16u;
        async_load16(lb +     0u, &Av[2 * row]);
        async_load16(lb +  4096u, &Av[2 * row + 1]);
        async_load16(lb +  8192u, &Bv[2 * row]);
        async_load16(lb + 12288u, &Bv[2 * row + 1]);
        wait_async_le4();
      } else {
        wait_async_0();
      }
      size_t row = (size_t)(tile0 + i) * TPB + t;
      float binf = __builtin_nontemporal_load(&Bin[row]);
      v4f a0 = smem[buf][0][t];
      v4f a1 = smem[buf][1][t];
      v4f b0 = smem[buf][2][t];
      v4f b1 = smem[buf][3][t];
      compute_store(pack8(a0, a1), pack8(b0, b1), (int)binf, outD, outB, row);
    }
  }

  // Tail rows (nRows not a multiple of TPB): direct streaming path.
  for (int row = nTiles * TPB + (int)blockIdx.x * TPB + t; row < nRows;
       row += (int)gridDim.x * TPB) {
    size_t r = (size_t)row;
    v4f a0 = __builtin_nontemporal_load(&Av[2 * r]);
    v4f a1 = __builtin_nontemporal_load(&Av[2 * r + 1]);
    v4f b0 = __builtin_nontemporal_load(&Bv[2 * r]);
    v4f b1 = __builtin_nontemporal_load(&Bv[2 * r + 1]);
    float binf = __builtin_nontemporal_load(&Bin[r]);
    compute_store(pack8(a0, a1), pack8(b0, b1), (int)binf, outD, outB, r);
  }
}

extern "C" void kernel_launch(void* const* d_in, const int* in_sizes, int n_in,
                              void* d_out, int out_size, void* d_ws, size_t ws_size,
                              hipStream_t stream) {
  const float* A   = (const float*)d_in[0];
  const float* B   = (const float*)d_in[1];
  const float* Bin = (const float*)d_in[2];
  float* out = (float*)d_out;

  const int nRows  = in_sizes[0] / 8;
  const int nTiles = nRows / TPB;
  int blocks = (nTiles + TILES_PER_BLOCK - 1) / TILES_PER_BLOCK;
  if (blocks < 1) blocks = 1;

  sub8_kernel<<<blocks, TPB, 0, stream>>>(A, B, Bin, out, nRows, nTiles);
}